// GroupAttention_73306501808135
// MI455X (gfx1250) — compile-verified
//
#include <hip/hip_runtime.h>
#include <hip/hip_bf16.h>
#include <math.h>

#define BS_ 2
#define H_  16
#define S_  1024
#define D_  1024
#define DK_ 64

typedef float v2f __attribute__((ext_vector_type(2)));
typedef float v8f __attribute__((ext_vector_type(8)));

// ---------------------------------------------------------------- LayerNorm
__global__ void ln_kernel(const float* __restrict__ ctx,
                          const float* __restrict__ lnw,
                          const float* __restrict__ lnb,
                          float* __restrict__ xln) {
  const int row = blockIdx.x;            // 0 .. bs*S-1
  const int t   = threadIdx.x;           // 256 threads, 4 elems each
  const float4 v = ((const float4*)(ctx + (size_t)row * D_))[t];
  float s  = v.x + v.y + v.z + v.w;
  float s2 = v.x*v.x + v.y*v.y + v.z*v.z + v.w*v.w;
  #pragma unroll
  for (int o = 16; o > 0; o >>= 1) {
    s  += __shfl_down(s,  o, 32);
    s2 += __shfl_down(s2, o, 32);
  }
  __shared__ float rs[8], rs2[8];
  __shared__ float mu_s, rstd_s;
  const int wid = t >> 5, lid = t & 31;
  if (lid == 0) { rs[wid] = s; rs2[wid] = s2; }
  __syncthreads();
  if (t == 0) {
    float a = 0.f, b = 0.f;
    #pragma unroll
    for (int i = 0; i < 8; ++i) { a += rs[i]; b += rs2[i]; }
    const float mu  = a / (float)D_;
    const float var = b / (float)D_ - mu * mu;   // biased variance (torch-style)
    mu_s = mu; rstd_s = rsqrtf(var + 1e-5f);
  }
  __syncthreads();
  const float mu = mu_s, rstd = rstd_s;
  const float4 w  = ((const float4*)lnw)[t];
  const float4 bb = ((const float4*)lnb)[t];
  float4 o;
  o.x = (v.x - mu) * rstd * w.x + bb.x;
  o.y = (v.y - mu) * rstd * w.y + bb.y;
  o.z = (v.z - mu) * rstd * w.z + bb.z;
  o.w = (v.w - mu) * rstd * w.w + bb.w;
  ((float4*)(xln + (size_t)row * D_))[t] = o;
}

// ------------------------------------------- Q/K projection via WMMA f32 16x16x4
// One wave computes a 16x64 Q tile and a 16x64 K tile for one (b,h,row-tile).
// A (16x4 f32): lane l holds M=l&15, K=4t+2*(l>>4)+{0,1}   (ISA 7.12.2)
// B (4x16 f32): lane l holds N=l&15, K=2*(l>>4)+{0,1}      (rows striped, half-lane split)
// C/D (16x16) : VGPR r, lane l -> M = r + 8*(l>>4), N = l&15
__global__ void proj_kernel(const float* __restrict__ xln,
                            const float* __restrict__ wq, const float* __restrict__ bq,
                            const float* __restrict__ wk, const float* __restrict__ bk,
                            float* __restrict__ qbuf, float* __restrict__ kbuf) {
  const int rt   = blockIdx.x;           // 0 .. bs*S/16 - 1
  const int h    = blockIdx.y;           // head
  const int lane = threadIdx.x;          // wave32
  const int lm   = lane & 15;
  const int lh   = lane >> 4;
  const int grow0 = rt * 16;             // flattened (b,s) row base; tiles never cross batch
  const int b  = grow0 / S_;
  const int s0 = grow0 % S_;

  // Hoist the A fragments for all 16 K-steps (this lane's 32 x values).
  const float* xrow = xln + (size_t)(grow0 + lm) * D_ + h * DK_;
  float ax[32];
  #pragma unroll
  for (int t = 0; t < 16; ++t) {
    ax[2*t]   = xrow[4*t + 2*lh];
    ax[2*t+1] = xrow[4*t + 2*lh + 1];
  }

  float* qout = qbuf + (((size_t)b * H_ + h) * S_ + s0) * DK_;
  float* kout = kbuf + (((size_t)b * H_ + h) * S_ + s0) * DK_;

  #pragma unroll
  for (int which = 0; which < 2; ++which) {
    const float* w   = which ? wk : wq;
    const float* bi  = which ? bk : bq;
    float*       out = which ? kout : qout;
    #pragma unroll
    for (int nt = 0; nt < 4; ++nt) {
      const int n0 = nt * 16;
      const float bias = bi[n0 + lm];    // bias depends only on N = lm
      v8f acc;
      #pragma unroll
      for (int r = 0; r < 8; ++r) acc[r] = bias;
      const float* wrow = w + (size_t)(n0 + lm) * DK_;   // B[k][n] = w[n][k]
      #pragma unroll
      for (int t = 0; t < 16; ++t) {
        v2f a;  a.x  = ax[2*t];               a.y  = ax[2*t+1];
        v2f bm; bm.x = wrow[4*t + 2*lh];      bm.y = wrow[4*t + 2*lh + 1];
        acc = __builtin_amdgcn_wmma_f32_16x16x4_f32(
            /*neg_a=*/false, a, /*neg_b=*/false, bm,
            /*c_mod=*/(short)0, acc, /*reuse_a=*/false, /*reuse_b=*/false);
      }
      #pragma unroll
      for (int r = 0; r < 8; ++r)
        out[(size_t)(r + 8*lh) * DK_ + n0 + lm] = acc[r];
    }
  }
}

// --------------------------------------- band scores: s_up[i]=q[i].k[i+1]/64 etc.
__global__ void band_kernel(const float* __restrict__ qbuf, const float* __restrict__ kbuf,
                            float* __restrict__ s_up, float* __restrict__ s_dn) {
  const int gw   = blockIdx.x * 8 + (threadIdx.x >> 5);  // = bh*S + i
  const int lane = threadIdx.x & 31;
  const int i    = gw % S_;
  const size_t base = (size_t)gw * DK_;
  const float q0 = qbuf[base + lane];
  const float q1 = qbuf[base + lane + 32];
  float pu = 0.f, pd = 0.f;
  if (i + 1 < S_) {
    const float* kn = kbuf + base + DK_;
    pu = q0 * kn[lane] + q1 * kn[lane + 32];
  }
  if (i > 0) {
    const float* kp = kbuf + base - DK_;
    pd = q0 * kp[lane] + q1 * kp[lane + 32];
  }
  #pragma unroll
  for (int o = 16; o > 0; o >>= 1) {
    pu += __shfl_down(pu, o, 32);
    pd += __shfl_down(pd, o, 32);
  }
  if (lane == 0) {
    s_up[gw] = (i + 1 < S_) ? pu * (1.0f / DK_) : -10000.0f;
    s_dn[gw] = (i > 0)      ? pd * (1.0f / DK_) : -10000.0f;
  }
}

// ------------------- 2-entry softmax (eos-masked) + symmetric geometric mean
__global__ void vsup_kernel(const float* __restrict__ s_up, const float* __restrict__ s_dn,
                            const unsigned char* __restrict__ eos,
                            float* __restrict__ vsup) {
  const int idx = blockIdx.x * blockDim.x + threadIdx.x;  // bh*S + i
  const int i  = idx % S_;
  const int bh = idx / S_;
  const int b  = bh / H_;
  if (i >= S_ - 1) return;
  const size_t mrow = (size_t)b * S_;
  auto eff = [&](int r, int c, float s) -> float {
    if (c < 0 || c >= S_) return -10000.0f;
    return eos[(mrow + r) * S_ + c] ? s : -10000.0f;
  };
  // softmax at row i
  const float su = eff(i, i + 1, s_up[idx]);
  const float sd = eff(i, i - 1, s_dn[idx]);
  const float m  = fmaxf(su, sd);
  const float eu = __expf(su - m), ed = __expf(sd - m);
  const float p_up = eu / (eu + ed);
  // softmax at row i+1, take the sub-diagonal prob p[i+1,i]
  const float su2 = eff(i + 1, i + 2, s_up[idx + 1]);
  const float sd2 = eff(i + 1, i,     s_dn[idx + 1]);
  const float m2  = fmaxf(su2, sd2);
  const float eu2 = __expf(su2 - m2), ed2 = __expf(sd2 - m2);
  const float p_dn = ed2 / (eu2 + ed2);
  vsup[idx] = sqrtf(p_up * p_dn + 1e-4f);
}

// ---------------- per-(b,h) prefix sum of L[i] = log(neibor[i,i+1] + 1e-9)
__global__ void scan_kernel(const float* __restrict__ prior, const float* __restrict__ vsup,
                            float* __restrict__ Cbuf) {
  const int bh = blockIdx.x;             // 0 .. bs*H-1
  const int t  = threadIdx.x;            // 1024
  __shared__ float arr[S_];
  float L = 0.f;
  if (t < S_ - 1) {
    const float pr = prior[((size_t)bh * S_ + t) * S_ + (t + 1)];
    const float nb = pr + (1.0f - pr) * vsup[bh * S_ + t];
    L = logf(nb + 1e-9f);
  }
  arr[t] = L;
  __syncthreads();
  for (int off = 1; off < S_; off <<= 1) {
    const float v = (t >= off) ? arr[t - off] : 0.f;
    __syncthreads();
    arr[t] += v;
    __syncthreads();
  }
  Cbuf[bh * S_ + t] = (t == 0) ? 0.f : arr[t - 1];   // exclusive prefix
}

// ------------------------------- bandwidth phase: write g and neibor, float4 wide
__global__ void out_kernel(const float* __restrict__ prior, const float* __restrict__ vsup,
                           const float* __restrict__ Cbuf,
                           float* __restrict__ g_out, float* __restrict__ nb_out) {
  const int row = blockIdx.x;            // bh*S + i
  const int i   = row % S_;
  const int bh  = row / S_;
  const int t   = threadIdx.x;           // 256 threads, 4 columns each
  const size_t rb = (size_t)row * S_;
  const float Ci     = Cbuf[bh * S_ + i];
  const float vs_i   = (i < S_ - 1) ? vsup[bh * S_ + i]     : 0.01f;
  const float vs_im1 = (i > 0)      ? vsup[bh * S_ + i - 1] : 0.01f;
  const int j0 = t * 4;
  const float4 pr = *(const float4*)(prior + rb + j0);
  const float4 Cj = *(const float4*)(Cbuf + (size_t)bh * S_ + j0);
  float4 nbv, gv;
  #pragma unroll
  for (int c = 0; c < 4; ++c) {
    const int j   = j0 + c;
    const float p = (&pr.x)[c];
    float v = 0.01f;                     // sqrt(0 + 1e-4): off-band and diagonal
    if      (j == i + 1) v = vs_i;
    else if (j == i - 1) v = vs_im1;
    const float n  = p + (1.0f - p) * v;
    const float cj = (&Cj.x)[c];
    const float gg = (j == i) ? n
                              : (__expf((j > i) ? (cj - Ci) : (Ci - cj)) + 1e-4f);
    (&nbv.x)[c] = n;
    (&gv.x)[c]  = gg;
  }
  *(float4*)(g_out  + rb + j0) = gv;
  *(float4*)(nb_out + rb + j0) = nbv;
}

// ---------------------------------------------------------------- launcher
extern "C" void kernel_launch(void* const* d_in, const int* in_sizes, int n_in,
                              void* d_out, int out_size, void* d_ws, size_t ws_size,
                              hipStream_t stream) {
  (void)in_sizes; (void)n_in; (void)out_size; (void)ws_size;
  const float*         ctx   = (const float*)d_in[0];
  const unsigned char* eos   = (const unsigned char*)d_in[1];  // jnp.bool_ -> 1 byte
  const float*         prior = (const float*)d_in[2];
  const float*         lnw   = (const float*)d_in[3];
  const float*         lnb   = (const float*)d_in[4];
  const float*         wk    = (const float*)d_in[5];
  const float*         bk    = (const float*)d_in[6];
  const float*         wq    = (const float*)d_in[7];
  const float*         bq    = (const float*)d_in[8];

  char* ws = (char*)d_ws;
  float* xln  = (float*)ws; ws += (size_t)BS_ * S_ * D_ * 4;
  float* qbuf = (float*)ws; ws += (size_t)BS_ * H_ * S_ * DK_ * 4;
  float* kbuf = (float*)ws; ws += (size_t)BS_ * H_ * S_ * DK_ * 4;
  float* s_up = (float*)ws; ws += (size_t)BS_ * H_ * S_ * 4;
  float* s_dn = (float*)ws; ws += (size_t)BS_ * H_ * S_ * 4;
  float* vsup = (float*)ws; ws += (size_t)BS_ * H_ * S_ * 4;
  float* Cbuf = (float*)ws; ws += (size_t)BS_ * H_ * S_ * 4;

  float* g_out  = (float*)d_out;
  float* nb_out = g_out + (size_t)BS_ * H_ * S_ * S_;

  ln_kernel<<<BS_ * S_, 256, 0, stream>>>(ctx, lnw, lnb, xln);

  dim3 pg(BS_ * S_ / 16, H_);
  proj_kernel<<<pg, 32, 0, stream>>>(xln, wq, bq, wk, bk, qbuf, kbuf);

  band_kernel<<<BS_ * H_ * S_ / 8, 256, 0, stream>>>(qbuf, kbuf, s_up, s_dn);

  vsup_kernel<<<BS_ * H_ * S_ / 256, 256, 0, stream>>>(s_up, s_dn, eos, vsup);

  scan_kernel<<<BS_ * H_, S_, 0, stream>>>(prior, vsup, Cbuf);

  out_kernel<<<BS_ * H_ * S_, 256, 0, stream>>>(prior, vsup, Cbuf, g_out, nb_out);
}